// GazeOnce360_40939628265601
// MI455X (gfx1250) — compile-verified
//
#include <hip/hip_runtime.h>
#include <hip/hip_bf16.h>

typedef float v2f __attribute__((ext_vector_type(2)));
typedef float v8f __attribute__((ext_vector_type(8)));

#define K_OUT   512
#define NBINS   2048
#define LN9     2.19722457733621938f
#define NMS_T   0.4f
#define IMG_SCALE 2048.0f

__device__ __forceinline__ int bin_of(float d) {
    // d in (ln9, ~16); monotone binning is all we need for radix-select
    float t = (d - LN9) * ((float)NBINS / 14.0f);
    int b = (int)t;
    b = b < 0 ? 0 : b;
    return b > (NBINS - 1) ? (NBINS - 1) : b;
}

// ---------------------------------------------------------------- zero scratch
__global__ void zero_kernel(int* p, int n) {
    int i = blockIdx.x * 256 + threadIdx.x;
    if (i < n) p[i] = 0;
}

// ------------------------------------------------- phase 1: threshold+compact
__global__ __launch_bounds__(256)
void score_compact_kernel(const float* __restrict__ conf,
                          float* __restrict__ candD, int* __restrict__ candIdx,
                          int* __restrict__ counter, int n, int cap) {
    int i = blockIdx.x * blockDim.x + threadIdx.x;
    int stride = gridDim.x * blockDim.x;
    int lane = threadIdx.x & 31;
    const float2* c2 = (const float2*)conf;
    for (; i < n; i += stride) {
        if (i + stride < n)
            __builtin_prefetch(&c2[i + stride], 0, 0);   // global_prefetch_b8
        float2 c = c2[i];
        float d = c.y - c.x;                      // logit; score = sigmoid(d)
        bool pred = d > LN9;                      // score > 0.9
        unsigned int m = (unsigned int)__ballot(pred);
        if (m) {
            int leader = __ffs(m) - 1;
            int base = 0;
            if (lane == leader) base = atomicAdd(counter, __popc(m));
            base = __shfl(base, leader, 32);
            if (pred) {
                int pos = base + __popc(m & ((1u << lane) - 1u));
                if (pos < cap) { candD[pos] = d; candIdx[pos] = i; }
            }
        }
    }
}

// ------------------------------------------------------- phase 2: histogram
__global__ __launch_bounds__(256)
void hist_kernel(const float* __restrict__ candD, const int* __restrict__ counter,
                 int* __restrict__ hist, int cap) {
    __shared__ int lh[NBINS];
    for (int i = threadIdx.x; i < NBINS; i += 256) lh[i] = 0;
    __syncthreads();
    int n = *counter; if (n > cap) n = cap;
    for (int i = blockIdx.x * blockDim.x + threadIdx.x; i < n;
         i += gridDim.x * blockDim.x)
        atomicAdd(&lh[bin_of(candD[i])], 1);
    __syncthreads();
    for (int i = threadIdx.x; i < NBINS; i += 256)
        if (lh[i]) atomicAdd(&hist[i], lh[i]);
}

// -------------------------------------- phase 3: select top-512 + bitonic sort
__global__ __launch_bounds__(512)
void select_sort_kernel(const float* __restrict__ candD, const int* __restrict__ candIdx,
                        const int* __restrict__ counter, const int* __restrict__ hist,
                        unsigned long long* __restrict__ selKeys, int cap) {
    __shared__ int shist[NBINS];
    __shared__ unsigned long long keys[K_OUT];
    __shared__ int sb, sneed, scnt, scntB, stotal;
    int tid = threadIdx.x;
    for (int i = tid; i < NBINS; i += 512) shist[i] = hist[i];
    keys[tid] = 0ull;
    if (tid == 0) { scnt = 0; scntB = 0; }
    __syncthreads();
    if (tid == 0) {
        int ncand = *counter; if (ncand > cap) ncand = cap;
        int target = ncand < K_OUT ? ncand : K_OUT;
        int suffix = 0, b = 0;
        for (int i = NBINS - 1; i >= 0; --i) {
            if (suffix + shist[i] >= target) { b = i; break; }
            suffix += shist[i];
        }
        sb = b; sneed = target - suffix; stotal = ncand;
    }
    __syncthreads();
    int b = sb, need = sneed, ncand = stotal;
    for (int p = tid; p < ncand; p += 512) {
        float d = candD[p];
        int bin = bin_of(d);
        if (bin >= b) {
            int slot = -1;
            if (bin > b) slot = atomicAdd(&scnt, 1);
            else { int r = atomicAdd(&scntB, 1); if (r < need) slot = atomicAdd(&scnt, 1); }
            if (slot >= 0 && slot < K_OUT) {
                unsigned int ord = __float_as_uint(d) ^ 0x80000000u; // d>0: flip sign bit
                unsigned int ii  = ~(unsigned int)candIdx[p];       // lower idx = bigger key
                keys[slot] = ((unsigned long long)ord << 32) | (unsigned long long)ii;
            }
        }
    }
    __syncthreads();
    // bitonic sort, descending by (score, -idx); padding keys==0 sink to bottom
    for (int k = 2; k <= K_OUT; k <<= 1) {
        for (int j = k >> 1; j > 0; j >>= 1) {
            int ixj = tid ^ j;
            if (ixj > tid) {
                bool desc = (tid & k) == 0;
                unsigned long long a = keys[tid], c = keys[ixj];
                bool sw = desc ? (a < c) : (a > c);
                if (sw) { keys[tid] = c; keys[ixj] = a; }
            }
            __syncthreads();
        }
    }
    selKeys[tid] = keys[tid];
}

// -------------------------------- phase 4: decode only the 512 survivors
__global__ __launch_bounds__(512)
void decode_kernel(const unsigned long long* __restrict__ selKeys,
                   const float* __restrict__ loc, const float* __restrict__ conf,
                   const float* __restrict__ priors,
                   float* __restrict__ gboxes, float* __restrict__ gscores) {
    int s = threadIdx.x;
    unsigned long long key = selKeys[s];
    if (key == 0ull) {
        gscores[s] = -1.0f;
        gboxes[s*4+0] = 0.f; gboxes[s*4+1] = 0.f; gboxes[s*4+2] = 0.f; gboxes[s*4+3] = 0.f;
        return;
    }
    int idx = (int)(~(unsigned int)(key & 0xFFFFFFFFull));
    float2 c = ((const float2*)conf)[idx];
    float d = c.y - c.x;
    float score = 1.0f / (1.0f + __expf(-d));
    float4 l = ((const float4*)loc)[idx];
    float4 p = ((const float4*)priors)[idx];
    float cx = p.x + l.x * 0.1f * p.z;
    float cy = p.y + l.y * 0.1f * p.w;
    float w  = p.z * __expf(l.z * 0.2f);
    float h  = p.w * __expf(l.w * 0.2f);
    float x1 = (cx - 0.5f * w) * IMG_SCALE;
    float y1 = (cy - 0.5f * h) * IMG_SCALE;
    float x2 = x1 + w * IMG_SCALE;
    float y2 = y1 + h * IMG_SCALE;
    gboxes[s*4+0] = x1; gboxes[s*4+1] = y1; gboxes[s*4+2] = x2; gboxes[s*4+3] = y2;
    gscores[s] = score;
}

// ---------------- phase 5: WMMA-tiled adjacency + bitmask NMS + output
__global__ __launch_bounds__(256)
void nms_kernel(const float* __restrict__ gboxes, const float* __restrict__ gscores,
                float* __restrict__ out) {
    __shared__ float bx1[K_OUT], by1[K_OUT], bx2[K_OUT], by2[K_OUT];
    __shared__ float tarea[K_OUT];              // NMS_T * area (folded threshold)
    __shared__ float fsc[K_OUT];
    __shared__ unsigned short adj[K_OUT][32];   // 512x512 bits = 32 KB
    __shared__ unsigned int keepw[16];
    int tid = threadIdx.x, lane = tid & 31, wave = tid >> 5;

    for (int s = tid; s < K_OUT; s += 256) {
        float x1 = gboxes[s*4+0], y1 = gboxes[s*4+1];
        float x2 = gboxes[s*4+2], y2 = gboxes[s*4+3];
        bx1[s] = x1; by1[s] = y1; bx2[s] = x2; by2[s] = y2;
        tarea[s] = NMS_T * (x2 - x1 + 1.0f) * (y2 - y1 + 1.0f);
        fsc[s] = gscores[s];
    }
    __syncthreads();

    // 32x32 tiles of 16x16; T*(area_i + area_j) via rank-2 outer product on WMMA.
    // iou > T  <=>  inter*(1+T) > T*(a_i + a_j - inter) + T*inter
    //          <=>  inter*(1+T) > D(m,n)   (division-free predicate)
    int li = lane & 15;
    bool lo = lane < 16;
    for (int t = wave; t < 1024; t += 8) {       // uniform per wave; EXEC all-1
        int ti = t >> 5, tj = t & 31;
        float ra = tarea[ti * 16 + li];
        float ca = tarea[tj * 16 + li];
        v2f a, bb;
        a.x  = lo ? ra   : 0.0f;  a.y  = lo ? 1.0f : 0.0f;   // A(m,0)=T*a_i, A(m,1)=1
        bb.x = lo ? 1.0f : 0.0f;  bb.y = lo ? ca   : 0.0f;   // B(0,n)=1, B(1,n)=T*a_j
        v8f cacc = {};
        v8f asum = __builtin_amdgcn_wmma_f32_16x16x4_f32(
            false, a, false, bb, (short)0, cacc, false, false);
        // D(m,n) = T*(a_i + a_j) ; C/D layout: lanes0-15 M=v, lanes16-31 M=v+8
        int n = tj * 16 + li;
        float jx1 = bx1[n], jy1 = by1[n], jx2 = bx2[n], jy2 = by2[n];
        int mbase = ti * 16 + (lo ? 0 : 8);
        for (int v = 0; v < 8; ++v) {
            int m = mbase + v;
            float xx1 = fmaxf(bx1[m], jx1), yy1 = fmaxf(by1[m], jy1);
            float xx2 = fminf(bx2[m], jx2), yy2 = fminf(by2[m], jy2);
            float w = fmaxf(xx2 - xx1 + 1.0f, 0.0f);
            float h = fmaxf(yy2 - yy1 + 1.0f, 0.0f);
            float inter = w * h;
            unsigned int bm = (unsigned int)__ballot(inter * (1.0f + NMS_T) > asum[v]);
            if (lane == 0) {
                adj[ti*16 + v][tj]     = (unsigned short)(bm & 0xFFFFu);
                adj[ti*16 + 8 + v][tj] = (unsigned short)(bm >> 16);
            }
        }
    }
    __syncthreads();

    if (tid < 16) {
        unsigned int kw = 0;
        for (int b = 0; b < 32; ++b)
            kw |= (fsc[tid*32 + b] > 0.0f ? 1u : 0u) << b;
        keepw[tid] = kw;
    }
    __syncthreads();

    if (wave == 0) {                 // serial dependence: 512 bitmask iterations
        for (int i = 0; i < K_OUT; ++i) {
            unsigned int kwi = keepw[i >> 5];
            if ((kwi >> (i & 31)) & 1u) {
                if (lane < 16) {
                    unsigned int m = (unsigned int)adj[i][2*lane] |
                                     ((unsigned int)adj[i][2*lane + 1] << 16);
                    int iw = i >> 5;
                    unsigned int gt = (lane > iw) ? 0xFFFFFFFFu
                                    : (lane < iw) ? 0u
                                    : ~((2u << (i & 31)) - 1u);   // bits strictly > i
                    keepw[lane] &= ~(m & gt);
                }
            }
        }
    }
    __syncthreads();

    const float inv = 1.0f / IMG_SCALE;
    for (int s = tid; s < K_OUT; s += 256) {
        bool kept = (keepw[s >> 5] >> (s & 31)) & 1u;
        out[s*5+0] = kept ? bx1[s] * inv : 0.0f;
        out[s*5+1] = kept ? by1[s] * inv : 0.0f;
        out[s*5+2] = kept ? bx2[s] * inv : 0.0f;
        out[s*5+3] = kept ? by2[s] * inv : 0.0f;
        out[s*5+4] = kept ? fsc[s] : 0.0f;
    }
}

extern "C" void kernel_launch(void* const* d_in, const int* in_sizes, int n_in,
                              void* d_out, int out_size, void* d_ws, size_t ws_size,
                              hipStream_t stream) {
    const float* loc    = (const float*)d_in[0];
    const float* conf   = (const float*)d_in[1];
    const float* priors = (const float*)d_in[2];
    float* out = (float*)d_out;
    int N = in_sizes[1] / 2;

    // ---- ws layout (bytes) ----
    char* ws = (char*)d_ws;
    int*   counter = (int*)(ws + 0);            // [0,4)
    int*   hist    = (int*)(ws + 1024);         // [1024, 9216)
    unsigned long long* selKeys = (unsigned long long*)(ws + 9216);   // [9216,13312)
    float* gboxes  = (float*)(ws + 13312);      // [13312, 21504)
    float* gscores = (float*)(ws + 21504);      // [21504, 23552)
    size_t fixed = 23552;
    long cap = 1024;
    if (ws_size > fixed + 4096) cap = (long)((ws_size - fixed) / 8);
    if (cap > 262144) cap = 262144;
    float* candD   = (float*)(ws + fixed);
    int*   candIdx = (int*)(ws + fixed + (size_t)cap * 4);

    // zero counter + histogram (graph-capture safe)
    zero_kernel<<<9, 256, 0, stream>>>((int*)ws, 2304);

    score_compact_kernel<<<2048, 256, 0, stream>>>(conf, candD, candIdx, counter, N, (int)cap);
    hist_kernel<<<512, 256, 0, stream>>>(candD, counter, hist, (int)cap);
    select_sort_kernel<<<1, 512, 0, stream>>>(candD, candIdx, counter, hist, selKeys, (int)cap);
    decode_kernel<<<1, 512, 0, stream>>>(selKeys, loc, conf, priors, gboxes, gscores);
    nms_kernel<<<1, 256, 0, stream>>>(gboxes, gscores, out);
}